// GraphNet_23871428231325
// MI455X (gfx1250) — compile-verified
//
#include <hip/hip_runtime.h>

#define Nn 20000
#define Ee 300000
#define Hh 256
#define Ll 4
#define Gg 64

typedef float v2f __attribute__((ext_vector_type(2)));
typedef float v8f __attribute__((ext_vector_type(8)));

// ---------------- degree / norm ----------------
__global__ __launch_bounds__(256) void deg_init_k(float* __restrict__ deg) {
    int i = blockIdx.x * 256 + threadIdx.x;
    if (i < Nn) deg[i] = 1.0f;   // self-loop
}

__global__ __launch_bounds__(256) void deg_edges_k(const int* __restrict__ dst,
                                                   float* __restrict__ deg) {
    int e = blockIdx.x * 256 + threadIdx.x;
    if (e < Ee) atomicAdd(&deg[dst[e]], 1.0f);
}

__global__ __launch_bounds__(256) void dinv_k(const float* __restrict__ deg,
                                              float* __restrict__ dinv) {
    int i = blockIdx.x * 256 + threadIdx.x;
    if (i < Nn) dinv[i] = rsqrtf(deg[i]);   // deg >= 1 guaranteed
}

// ---------------- embedding gather ----------------
__global__ __launch_bounds__(256) void gather_k(const int* __restrict__ atoms,
                                                const float* __restrict__ embed,
                                                float* __restrict__ x) {
    int idx4 = blockIdx.x * 256 + threadIdx.x;       // float4 index
    if (idx4 >= Nn * (Hh / 4)) return;
    int i  = idx4 >> 6;          // node
    int j4 = idx4 & 63;          // float4 within row
    const float4* src = (const float4*)(embed + (size_t)atoms[i] * Hh);
    ((float4*)(x + (size_t)i * Hh))[j4] = src[j4];
}

// ---------------- GEMM: h = x @ W  (N x 256 @ 256 x 256), fp32 WMMA ----------------
__global__ __launch_bounds__(256) void gemm_wmma_k(const float* __restrict__ X,
                                                   const float* __restrict__ W,
                                                   float* __restrict__ Hout) {
    __shared__ float xs[16 * 260];                   // padded stride: 0 bank conflicts
    const int tid  = threadIdx.x;
    const int wave = tid >> 5;
    const int lane = tid & 31;
    const int mbase = blockIdx.x * 16;

    // stage 16x256 x-tile into LDS, coalesced (row j per iteration)
    #pragma unroll
    for (int it = 0; it < 16; ++it) {
        int r = it;
        int c = tid;
        xs[r * 260 + c] = X[(size_t)(mbase + r) * Hh + c];
    }
    __syncthreads();

    const int m    = lane & 15;              // A: row within M-tile
    const int koff = (lane >> 4) << 1;       // A/B: K sub-offset (0 or 2)
    const int n    = lane & 15;              // B/C/D: column within N-tile
    const int half = lane >> 4;              // C/D: M half select
    const int nb0  = wave * 32;              // this wave's first N-tile
    const int nb1  = nb0 + 16;               // second N-tile

    v8f c0 = {};
    v8f c1 = {};

    for (int k0 = 0; k0 < Hh; k0 += 4) {
        v2f a;
        a.x = xs[m * 260 + k0 + koff];
        a.y = xs[m * 260 + k0 + koff + 1];
        const float* w0 = W + (size_t)(k0 + koff) * Hh;
        v2f b0, b1;
        b0.x = w0[nb0 + n];
        b0.y = w0[Hh + nb0 + n];
        b1.x = w0[nb1 + n];
        b1.y = w0[Hh + nb1 + n];
        c0 = __builtin_amdgcn_wmma_f32_16x16x4_f32(false, a, false, b0,
                                                   (short)0, c0, false, false);
        c1 = __builtin_amdgcn_wmma_f32_16x16x4_f32(false, a, false, b1,
                                                   (short)0, c1, false, false);
    }

    // C/D layout: VGPR r, lane-half h -> M = r + 8*h ; N = lane&15
    #pragma unroll
    for (int r = 0; r < 8; ++r) {
        int gm = mbase + r + 8 * half;
        Hout[(size_t)gm * Hh + nb0 + n] = c0[r];
        Hout[(size_t)gm * Hh + nb1 + n] = c1[r];
    }
}

// ---------------- agg init: bias + self-loop term ----------------
__global__ __launch_bounds__(256) void agg_init_k(const float* __restrict__ Hm,
                                                  const float* __restrict__ dinv,
                                                  const float* __restrict__ bias,
                                                  float* __restrict__ agg) {
    int idx4 = blockIdx.x * 256 + threadIdx.x;
    if (idx4 >= Nn * (Hh / 4)) return;
    int i  = idx4 >> 6;
    int j4 = idx4 & 63;
    float di = dinv[i];
    float nrm = di * di;                 // self-loop norm
    float4 hv = ((const float4*)(Hm + (size_t)i * Hh))[j4];
    float4 bv = ((const float4*)bias)[j4];
    float4 r;
    r.x = bv.x + hv.x * nrm;
    r.y = bv.y + hv.y * nrm;
    r.z = bv.z + hv.z * nrm;
    r.w = bv.w + hv.w * nrm;
    ((float4*)(agg + (size_t)i * Hh))[j4] = r;
}

// ---------------- edge scatter: agg[dst] += h[src] * norm  (wave per edge) ----------------
__global__ __launch_bounds__(256) void edge_scatter_k(const int* __restrict__ src,
                                                      const int* __restrict__ dst,
                                                      const float* __restrict__ dinv,
                                                      const float* __restrict__ Hm,
                                                      float* __restrict__ agg) {
    int e = blockIdx.x * 8 + (threadIdx.x >> 5);
    if (e >= Ee) return;
    int lane = threadIdx.x & 31;
    int s = src[e];
    int d = dst[e];
    float nrm = dinv[s] * dinv[d];
    const float4* hp = (const float4*)(Hm + (size_t)s * Hh);
    float* ap = agg + (size_t)d * Hh + lane * 8;
    float4 v0 = hp[lane * 2];
    float4 v1 = hp[lane * 2 + 1];
    atomicAdd(ap + 0, v0.x * nrm);
    atomicAdd(ap + 1, v0.y * nrm);
    atomicAdd(ap + 2, v0.z * nrm);
    atomicAdd(ap + 3, v0.w * nrm);
    atomicAdd(ap + 4, v1.x * nrm);
    atomicAdd(ap + 5, v1.y * nrm);
    atomicAdd(ap + 6, v1.z * nrm);
    atomicAdd(ap + 7, v1.w * nrm);
}

// ---------------- residual + relu: x = relu(x + agg) ----------------
__global__ __launch_bounds__(256) void relu_upd_k(float* __restrict__ x,
                                                  const float* __restrict__ agg) {
    int idx4 = blockIdx.x * 256 + threadIdx.x;
    if (idx4 >= Nn * (Hh / 4)) return;
    float4 xv = ((const float4*)x)[idx4];
    float4 av = ((const float4*)agg)[idx4];
    float4 r;
    r.x = fmaxf(xv.x + av.x, 0.0f);
    r.y = fmaxf(xv.y + av.y, 0.0f);
    r.z = fmaxf(xv.z + av.z, 0.0f);
    r.w = fmaxf(xv.w + av.w, 0.0f);
    ((float4*)x)[idx4] = r;
}

// ---------------- pooling ----------------
__global__ __launch_bounds__(256) void pool_zero_k(float* __restrict__ pooled,
                                                   float* __restrict__ counts) {
    int i = blockIdx.x * 256 + threadIdx.x;
    if (i < Gg * Hh) pooled[i] = 0.0f;
    if (i < Gg) counts[i] = 0.0f;
}

__global__ __launch_bounds__(256) void count_acc_k(const int* __restrict__ batch,
                                                   float* __restrict__ counts) {
    int i = blockIdx.x * 256 + threadIdx.x;
    if (i < Nn) atomicAdd(&counts[batch[i]], 1.0f);
}

__global__ __launch_bounds__(256) void pool_acc_k(const int* __restrict__ batch,
                                                  const float* __restrict__ x,
                                                  float* __restrict__ pooled) {
    int idx = blockIdx.x * 256 + threadIdx.x;
    if (idx >= Nn * Hh) return;
    int i = idx >> 8;
    int j = idx & 255;
    atomicAdd(&pooled[(size_t)batch[i] * Hh + j], x[idx]);
}

__global__ __launch_bounds__(64) void final_out_k(const float* __restrict__ pooled,
                                                  const float* __restrict__ counts,
                                                  const float* __restrict__ fcw,
                                                  const float* __restrict__ fcb,
                                                  float* __restrict__ out) {
    int g = threadIdx.x;
    if (g >= Gg) return;
    float inv = 1.0f / fmaxf(counts[g], 1.0f);
    float acc = 0.0f;
    const float* pr = pooled + (size_t)g * Hh;
    #pragma unroll 8
    for (int j = 0; j < Hh; ++j) acc += pr[j] * fcw[j];
    out[g] = acc * inv + fcb[0];
}

// ---------------- launch ----------------
extern "C" void kernel_launch(void* const* d_in, const int* in_sizes, int n_in,
                              void* d_out, int out_size, void* d_ws, size_t ws_size,
                              hipStream_t stream) {
    const int*   atoms = (const int*)d_in[0];
    const int*   eidx  = (const int*)d_in[1];   // [2, E]: row0=src, row1=dst
    const int*   batch = (const int*)d_in[2];
    const float* embed = (const float*)d_in[3]; // [V, H]
    const float* Ws    = (const float*)d_in[4]; // [L, H, H]
    const float* bs    = (const float*)d_in[5]; // [L, H]
    const float* fcw   = (const float*)d_in[6]; // [H]
    const float* fcb   = (const float*)d_in[7]; // [1]
    float* out = (float*)d_out;

    const int* esrc = eidx;
    const int* edst = eidx + Ee;

    float* ws     = (float*)d_ws;
    float* deg    = ws;                   // N
    float* dinv   = deg  + Nn;            // N
    float* x      = dinv + Nn;            // N*H
    float* h      = x    + (size_t)Nn * Hh;
    float* agg    = h    + (size_t)Nn * Hh;
    float* pooled = agg  + (size_t)Nn * Hh;   // G*H
    float* counts = pooled + Gg * Hh;         // G

    const int nb_nodes = (Nn + 255) / 256;
    const int nb_edges = (Ee + 255) / 256;
    const int nb_vec4  = (Nn * (Hh / 4) + 255) / 256;   // float4 elementwise
    const int nb_elem  = (Nn * Hh + 255) / 256;

    // degrees + normalization
    deg_init_k<<<nb_nodes, 256, 0, stream>>>(deg);
    deg_edges_k<<<nb_edges, 256, 0, stream>>>(edst, deg);
    dinv_k<<<nb_nodes, 256, 0, stream>>>(deg, dinv);

    // embedding gather
    gather_k<<<nb_vec4, 256, 0, stream>>>(atoms, embed, x);

    // layers
    for (int l = 0; l < Ll; ++l) {
        const float* Wl = Ws + (size_t)l * Hh * Hh;
        const float* bl = bs + (size_t)l * Hh;
        gemm_wmma_k<<<Nn / 16, 256, 0, stream>>>(x, Wl, h);
        agg_init_k<<<nb_vec4, 256, 0, stream>>>(h, dinv, bl, agg);
        edge_scatter_k<<<(Ee + 7) / 8, 256, 0, stream>>>(esrc, edst, dinv, h, agg);
        relu_upd_k<<<nb_vec4, 256, 0, stream>>>(x, agg);
    }

    // mean pool + FC
    pool_zero_k<<<(Gg * Hh + 255) / 256, 256, 0, stream>>>(pooled, counts);
    count_acc_k<<<nb_nodes, 256, 0, stream>>>(batch, counts);
    pool_acc_k<<<nb_elem, 256, 0, stream>>>(batch, x, pooled);
    final_out_k<<<1, 64, 0, stream>>>(pooled, counts, fcw, fcb, out);
}